// RQspline_59940563583739
// MI455X (gfx1250) — compile-verified
//
#include <hip/hip_runtime.h>

#define NDIM     256
#define NKNOT    32
#define NPAD     33          // 33 coprime with 64 banks -> conflict-free LDS tables
#define NTHREADS 256
#define NBLOCKS  1024

__global__ __launch_bounds__(NTHREADS) void rqspline_fwd_kernel(
    const float* __restrict__ x,
    const float* __restrict__ x0,
    const float* __restrict__ y0,
    const float* __restrict__ logdx,
    const float* __restrict__ logdy,
    const float* __restrict__ logderiv,
    float* __restrict__ yout,
    float* __restrict__ ldout,
    int nrows)
{
    __shared__ float s_xx[NDIM * NPAD];   // knot x positions, [d*33 + k]
    __shared__ float s_yy[NDIM * NPAD];   // knot y positions
    __shared__ float s_dl[NDIM * NPAD];   // derivatives delta = exp(logderiv)

    const int t = threadIdx.x;            // thread t owns dim d == t

    // ---------- Phase 0: CDNA5 async DMA: logderiv (raw bits) -> LDS ----------
    // Per-lane LDS destination address performs the (d,k) -> padded-row scatter
    // for free while the global side stays perfectly coalesced (e = j*256 + t).
    {
        #pragma unroll
        for (int j = 0; j < NKNOT; ++j) {
            unsigned e   = (unsigned)(j * NTHREADS + t);
            unsigned dd  = e >> 5;
            unsigned kk  = e & 31u;
            unsigned lds = (unsigned)(size_t)&s_dl[dd * NPAD + kk]; // low 32b of flat = LDS offset
            unsigned off = e * 4u;
            asm volatile("global_load_async_to_lds_b32 %0, %1, %2"
                         :: "v"(lds), "v"(off), "s"(logderiv)
                         : "memory");
        }
        asm volatile("s_wait_asynccnt 0" ::: "memory");
    }

    // ---------- Phase 1: per-dim cumulative knot tables (thread t = dim t) ----------
    {
        float cx = x0[t];
        float cy = y0[t];
        s_xx[t * NPAD + 0] = cx;
        s_yy[t * NPAD + 0] = cy;
        #pragma unroll 1
        for (int k = 1; k < NKNOT; ++k) {
            cx += __expf(logdx[t * (NKNOT - 1) + (k - 1)]);
            cy += __expf(logdy[t * (NKNOT - 1) + (k - 1)]);
            s_xx[t * NPAD + k] = cx;
            s_yy[t * NPAD + k] = cy;
        }
    }
    __syncthreads();                       // async s_dl writes from all waves visible

    // delta = exp(logderiv), in place (thread t exponentiates its own row)
    #pragma unroll 1
    for (int k = 0; k < NKNOT; ++k)
        s_dl[t * NPAD + k] = __expf(s_dl[t * NPAD + k]);
    __syncthreads();

    // ---------- Phase 2: stream rows ----------
    const int d = t;
    const float* __restrict__ xxp = &s_xx[d * NPAD];
    const float* __restrict__ yyp = &s_yy[d * NPAD];
    const float* __restrict__ dlp = &s_dl[d * NPAD];

    // Edge-case constants for this dim, hoisted once.
    const float xx0 = xxp[0],         xxL = xxp[NKNOT - 1];
    const float yy0 = yyp[0],         yyL = yyp[NKNOT - 1];
    const float dl0 = dlp[0],         dlL = dlp[NKNOT - 1];
    const float lr0 = logderiv[d * NKNOT + 0];
    const float lrL = logderiv[d * NKNOT + (NKNOT - 1)];

    const int chunk = (nrows + (int)gridDim.x - 1) / (int)gridDim.x;
    int rbeg = (int)blockIdx.x * chunk;
    int rend = rbeg + chunk;
    if (rend > nrows) rend = nrows;

    for (int row = rbeg; row < rend; ++row) {
        if (row + 16 < rend)              // global_prefetch_b8, 16 rows (16 KB) ahead
            __builtin_prefetch(&x[(size_t)(row + 16) * NDIM + d], 0, 0);

        const float xt = x[(size_t)row * NDIM + d];

        // Branchless binary search: idx = #{knots[0..30] < xt}  in [0, 31]  (side='left')
        int idx = 0;
        #pragma unroll
        for (int step = 16; step >= 1; step >>= 1) {
            int cand = idx + step;
            float kv = xxp[cand - 1];
            idx = (kv < xt) ? cand : idx;
        }
        int k = idx < 1 ? 1 : (idx > NKNOT - 1 ? NKNOT - 1 : idx);

        // Paired gathers (adjacent k-1/k -> ds_load_2addr candidates)
        const float xkm = xxp[k - 1], xk = xxp[k];
        const float ykm = yyp[k - 1], yk = yyp[k];
        const float dkm = dlp[k - 1], dk = dlp[k];

        const float dx   = xk - xkm;
        const float rdx  = __builtin_amdgcn_rcpf(dx);
        const float sdy  = yk - ykm;
        const float s    = sdy * rdx;
        const float xi   = (xt - xkm) * rdx;
        const float omxi = 1.0f - xi;
        const float xi1  = xi * omxi;
        const float xi2  = xi * xi;

        float tden = dk + dkm;
        tden = __builtin_fmaf(-2.0f, s, tden);
        const float denom = __builtin_fmaf(tden, xi1, s);
        const float rden  = __builtin_amdgcn_rcpf(denom);

        const float numy  = __builtin_fmaf(dkm, xi1, s * xi2);
        const float y_int = __builtin_fmaf(sdy * rden, numy, ykm);

        float num2 = __builtin_fmaf(dk, xi2, 2.0f * s * xi1);
        num2 = __builtin_fmaf(dkm * omxi, omxi, num2);
        const float ratio  = s * rden;
        // 2*log(s) + log(num2) - 2*log(denom) == log((s/denom)^2 * num2)
        const float ld_int = __logf(ratio * ratio * num2);

        // Linear extrapolation outside the knot range
        const float y_lo = __builtin_fmaf(xt - xx0, dl0, yy0);
        const float y_hi = __builtin_fmaf(xt - xxL, dlL, yyL);

        const float y  = (idx == 0) ? y_lo : ((idx == NKNOT) ? y_hi : y_int);
        const float ld = (idx == 0) ? lr0  : ((idx == NKNOT) ? lrL  : ld_int);

        __builtin_nontemporal_store(y,  &yout [(size_t)row * NDIM + d]);
        __builtin_nontemporal_store(ld, &ldout[(size_t)row * NDIM + d]);
    }
}

extern "C" void kernel_launch(void* const* d_in, const int* in_sizes, int n_in,
                              void* d_out, int out_size, void* d_ws, size_t ws_size,
                              hipStream_t stream) {
    const float* x        = (const float*)d_in[0];
    const float* x0       = (const float*)d_in[1];
    const float* y0       = (const float*)d_in[2];
    const float* logdx    = (const float*)d_in[3];
    const float* logdy    = (const float*)d_in[4];
    const float* logderiv = (const float*)d_in[5];

    const int nrows = in_sizes[0] / NDIM;           // 131072
    float* yout  = (float*)d_out;                   // y  : nrows*NDIM
    float* ldout = yout + (size_t)nrows * NDIM;     // ld : nrows*NDIM

    dim3 grid(NBLOCKS), block(NTHREADS);
    hipLaunchKernelGGL(rqspline_fwd_kernel, grid, block, 0, stream,
                       x, x0, y0, logdx, logdy, logderiv, yout, ldout, nrows);
}